// GCNLayer_33182917328985
// MI455X (gfx1250) — compile-verified
//
#include <hip/hip_runtime.h>

#define D_FEAT 128

typedef __attribute__((ext_vector_type(2))) float v2f;
typedef __attribute__((ext_vector_type(8))) float v8f;

// ---------------------------------------------------------------------------
// Kernel 1: zero the aggregation accumulator (graph-capture-safe, no memset).
// ---------------------------------------------------------------------------
__global__ void gcn_zero_kernel(float* __restrict__ h, int n) {
    int i = blockIdx.x * blockDim.x + threadIdx.x;
    if (i < n) h[i] = 0.0f;
}

// ---------------------------------------------------------------------------
// Kernel 2: edge scatter.  One wave32 per edge: each lane loads 4 consecutive
// floats of the source row (GLOBAL_LOAD_B128) and atomically accumulates into
// the destination row (GLOBAL_ATOMIC_ADD_F32, no return -> STOREcnt).
// x (5.12 MB) and h (5.12 MB) both live in the 192 MB L2, so the random
// gather/scatter never goes to HBM.
// ---------------------------------------------------------------------------
__global__ void gcn_scatter_kernel(const float* __restrict__ x,
                                   const int* __restrict__ src,
                                   const int* __restrict__ dst,
                                   float* __restrict__ h,
                                   int n_edges) {
    int t    = blockIdx.x * blockDim.x + threadIdx.x;
    int e    = t >> 5;
    int lane = t & 31;
    if (e >= n_edges) return;

    int s = src[e];
    int d = dst[e];

    const float4 v = *(const float4*)(x + (size_t)s * D_FEAT + lane * 4);
    float* hp = h + (size_t)d * D_FEAT + lane * 4;
    atomicAdd(hp + 0, v.x);
    atomicAdd(hp + 1, v.y);
    atomicAdd(hp + 2, v.z);
    atomicAdd(hp + 3, v.w);
}

// ---------------------------------------------------------------------------
// Kernel 3: out = h @ W^T + b using V_WMMA_F32_16X16X4_F32 (full fp32).
// One wave per 16x16 output tile; 10000 = 625*16 so no M tail.
//
// A (16x4 f32) lane layout: lane L -> row M=L&15; VGPR0/1 hold K={0,1}
// (lanes 0-15) or K={2,3} (lanes 16-31)  => kb = (lane>>4)*2.
// B (4x16 f32) mirrors A with N=lane&15.  Our B[k][n] = W[n][k] (out=h@W^T),
// so the B fetch reads row n of W — same addressing pattern as A.
// C/D (16x16 f32, 8 VGPRs): VGPR j -> row (lane<16 ? j : j+8), col lane&15.
// ---------------------------------------------------------------------------
__global__ void gcn_wmma_gemm_kernel(const float* __restrict__ h,
                                     const float* __restrict__ W,
                                     const float* __restrict__ bias,
                                     float* __restrict__ out,
                                     int n_tiles_m) {
    int wave = (blockIdx.x * blockDim.x + threadIdx.x) >> 5;
    int lane = threadIdx.x & 31;
    int total_tiles = n_tiles_m * (D_FEAT / 16);   // 625 * 8 = 5000
    if (wave >= total_tiles) return;

    int tm = wave >> 3;          // M tile index (D_FEAT/16 == 8 N tiles)
    int tn = wave & 7;           // N tile index

    int m  = lane & 15;          // row (A) / col (B) handled by this lane
    int kb = (lane >> 4) << 1;   // 0 for lanes 0-15, 2 for lanes 16-31

    const float* arow = h + (size_t)(tm * 16 + m) * D_FEAT;  // A: h rows
    const float* brow = W + (size_t)(tn * 16 + m) * D_FEAT;  // B: W rows (W^T cols)

    v8f c = {};
#pragma unroll
    for (int k0 = 0; k0 < D_FEAT; k0 += 4) {
        v2f a, b;
        // even base offset -> these lower to 8-byte (b64) loads
        a.x = arow[k0 + kb];
        a.y = arow[k0 + kb + 1];
        b.x = brow[k0 + kb];
        b.y = brow[k0 + kb + 1];
        // (neg_a, A, neg_b, B, c_mod, C, reuse_a, reuse_b)
        c = __builtin_amdgcn_wmma_f32_16x16x4_f32(
                false, a, false, b, (short)0, c, false, false);
    }

    float bval = bias[tn * 16 + m];              // bias indexed by output col
    int row_base = tm * 16 + ((lane >> 4) << 3); // lanes 16-31 -> rows +8
    float* op = out + (size_t)row_base * D_FEAT + tn * 16 + m;
#pragma unroll
    for (int j = 0; j < 8; ++j) {
        op[(size_t)j * D_FEAT] = c[j] + bval;
    }
}

// ---------------------------------------------------------------------------
// Launcher
// ---------------------------------------------------------------------------
extern "C" void kernel_launch(void* const* d_in, const int* in_sizes, int n_in,
                              void* d_out, int out_size, void* d_ws, size_t ws_size,
                              hipStream_t stream) {
    const float* x   = (const float*)d_in[0];   // [N, 128]
    const int*   src = (const int*)d_in[1];     // [E] (JAX demotes int64->int32)
    const int*   dst = (const int*)d_in[2];     // [E]
    const float* W   = (const float*)d_in[3];   // [128, 128]
    const float* b   = (const float*)d_in[4];   // [128]
    float*       out = (float*)d_out;           // [N, 128]

    const int n_nodes = in_sizes[0] / D_FEAT;   // 10000
    const int n_edges = in_sizes[1];            // 640000

    float* h = (float*)d_ws;                    // [N, 128] accumulator (5.12 MB)

    // 1) zero accumulator
    {
        int n = n_nodes * D_FEAT;
        int blk = 256;
        int grid = (n + blk - 1) / blk;
        gcn_zero_kernel<<<grid, blk, 0, stream>>>(h, n);
    }

    // 2) edge scatter: one wave32 per edge
    {
        long long threads = (long long)n_edges * 32;
        int blk = 256;
        long long grid = (threads + blk - 1) / blk;
        gcn_scatter_kernel<<<(int)grid, blk, 0, stream>>>(x, src, dst, h, n_edges);
    }

    // 3) WMMA fp32 GEMM + bias: one wave per 16x16 tile, 8 waves per block
    {
        int n_tiles_m = n_nodes / 16;                 // 625 (exact)
        int total_tiles = n_tiles_m * (D_FEAT / 16);  // 5000 waves
        int blk = 256;                                // 8 waves/block
        int grid = (total_tiles * 32 + blk - 1) / blk;
        gcn_wmma_gemm_kernel<<<grid, blk, 0, stream>>>(h, W, b, out, n_tiles_m);
    }
}